// PatchIndicatorGrid_14104672600555
// MI455X (gfx1250) — compile-verified
//
#include <hip/hip_runtime.h>

// Trilinear interpolation of a (128,128,128,8) f32 grid at N points.
// Gather/L2-bound: grid (64MB) is L2-resident on MI455X (192MB L2);
// coords/output are streamed with non-temporal hints to protect L2 residency.

typedef __attribute__((ext_vector_type(4))) float f32x4;

__device__ __forceinline__ f32x4 ld4(const float* __restrict__ p) {
    return *(const f32x4* __restrict__)p;
}

__device__ __forceinline__ f32x4 lerp4(f32x4 a, f32x4 b, float w) {
    // fma(w, b-a, a) — contraction fuses this into v_fma ops
    return a + w * (b - a);
}

__global__ __launch_bounds__(256)
void trilerp_grid_kernel(const float* __restrict__ coords,
                         const float* __restrict__ grid,
                         const float* __restrict__ bbox_min,
                         const float* __restrict__ bbox_max,
                         float* __restrict__ out,
                         int n)
{
    const int i = blockIdx.x * 256 + threadIdx.x;
    if (i >= n) return;

    // Uniform bbox -> scalar loads; 1/scale via v_rcp_f32 (1-ulp) instead of
    // the full IEEE div_scale/div_fmas chain (uniform divisor; exact for the
    // unit bbox used here).
    const float bx = bbox_min[0], by = bbox_min[1], bz = bbox_min[2];
    const float eps = 1e-6f;
    const float rsx = __builtin_amdgcn_rcpf(fmaxf(bbox_max[0] - bx, eps));
    const float rsy = __builtin_amdgcn_rcpf(fmaxf(bbox_max[1] - by, eps));
    const float rsz = __builtin_amdgcn_rcpf(fmaxf(bbox_max[2] - bz, eps));

    // Streaming coord read (12B/lane, contiguous across wave) — non-temporal.
    const float* __restrict__ cp = coords + (size_t)3 * (size_t)i;
    const float cx = __builtin_nontemporal_load(cp + 0);
    const float cy = __builtin_nontemporal_load(cp + 1);
    const float cz = __builtin_nontemporal_load(cp + 2);

    // normalized = clip((c - bmin) * rscale, 0, 1); pos = normalized * 127
    const float nx = fminf(fmaxf((cx - bx) * rsx, 0.0f), 1.0f);
    const float ny = fminf(fmaxf((cy - by) * rsy, 0.0f), 1.0f);
    const float nz = fminf(fmaxf((cz - bz) * rsz, 0.0f), 1.0f);
    const float px = nx * 127.0f;
    const float py = ny * 127.0f;
    const float pz = nz * 127.0f;

    // floor == truncate (pos >= 0)
    const int x0 = (int)px, y0 = (int)py, z0 = (int)pz;
    const int x1 = min(x0 + 1, 127);
    const int y1 = min(y0 + 1, 127);
    const int z1 = min(z0 + 1, 127);

    const float wx = fminf(fmaxf(px - (float)x0, 0.0f), 1.0f);
    const float wy = fminf(fmaxf(py - (float)y0, 0.0f), 1.0f);
    const float wz = fminf(fmaxf(pz - (float)z0, 0.0f), 1.0f);

    // Flat strides (floats): x: 1<<17, y: 1<<10, z: 1<<3. Unsigned offsets so
    // the compiler can use SADDR(base)+voffset global addressing; the +4-float
    // channel offset folds into the instruction immediate (offset:16).
    const unsigned ux0 = (unsigned)x0 << 17, ux1 = (unsigned)x1 << 17;
    const unsigned uy0 = (unsigned)y0 << 10, uy1 = (unsigned)y1 << 10;
    const unsigned uz0 = (unsigned)z0 << 3,  uz1 = (unsigned)z1 << 3;

    // For each (x,y) corner: gather z0/z1 channel vectors (2x b128 each),
    // lerp along z immediately to cut live registers.
    f32x4 r00a, r00b, r10a, r10b, r01a, r01b, r11a, r11b;
    {
        const unsigned b = ux0 + uy0;
        const float* p0 = grid + (b + uz0);
        const float* p1 = grid + (b + uz1);
        r00a = lerp4(ld4(p0),     ld4(p1),     wz);
        r00b = lerp4(ld4(p0 + 4), ld4(p1 + 4), wz);
    }
    {
        const unsigned b = ux1 + uy0;
        const float* p0 = grid + (b + uz0);
        const float* p1 = grid + (b + uz1);
        r10a = lerp4(ld4(p0),     ld4(p1),     wz);
        r10b = lerp4(ld4(p0 + 4), ld4(p1 + 4), wz);
    }
    {
        const unsigned b = ux0 + uy1;
        const float* p0 = grid + (b + uz0);
        const float* p1 = grid + (b + uz1);
        r01a = lerp4(ld4(p0),     ld4(p1),     wz);
        r01b = lerp4(ld4(p0 + 4), ld4(p1 + 4), wz);
    }
    {
        const unsigned b = ux1 + uy1;
        const float* p0 = grid + (b + uz0);
        const float* p1 = grid + (b + uz1);
        r11a = lerp4(ld4(p0),     ld4(p1),     wz);
        r11b = lerp4(ld4(p0 + 4), ld4(p1 + 4), wz);
    }

    // Bilinear combine in x then y.
    const f32x4 ta = lerp4(r00a, r10a, wx);
    const f32x4 tb = lerp4(r00b, r10b, wx);
    const f32x4 ua = lerp4(r01a, r11a, wx);
    const f32x4 ub = lerp4(r01b, r11b, wx);
    const f32x4 oa = lerp4(ta, ua, wy);
    const f32x4 ob = lerp4(tb, ub, wy);

    // Streaming 32B output per lane — non-temporal b128 stores.
    f32x4* __restrict__ op = (f32x4* __restrict__)(out + (size_t)8 * (size_t)i);
    __builtin_nontemporal_store(oa, op + 0);
    __builtin_nontemporal_store(ob, op + 1);
}

extern "C" void kernel_launch(void* const* d_in, const int* in_sizes, int n_in,
                              void* d_out, int out_size, void* d_ws, size_t ws_size,
                              hipStream_t stream)
{
    const float* coords   = (const float*)d_in[0];  // (N,3) f32
    const float* grid     = (const float*)d_in[1];  // (128,128,128,8) f32
    const float* bbox_min = (const float*)d_in[2];  // (3,) f32
    const float* bbox_max = (const float*)d_in[3];  // (3,) f32
    float* out            = (float*)d_out;          // (N,8) f32

    const int n = in_sizes[0] / 3;                  // 2,000,000 points
    const int block = 256;                          // 8 wave32 waves
    const int gridDim = (n + block - 1) / block;

    trilerp_grid_kernel<<<gridDim, block, 0, stream>>>(
        coords, grid, bbox_min, bbox_max, out, n);
}